// QuantumKernelRegressor_65481071396992
// MI455X (gfx1250) — compile-verified
//
#include <hip/hip_runtime.h>
#include <math.h>

// QuantumKernelRegressor on gfx1250 (MI455X), wave32.
//
// K[i,j] = prod_f cos^2((X[i,f]-S[j,f])/2)
//        = 2^-64 * prod_f (1 + cosX*cosS + sinX*sinS)     [trig identity]
// out[i] = sum_j K[i,j] * W[j] + b
//
// - sincos hoisted: (B+(B/16)*S)*F evals instead of B*S*F (~16x fewer TRANS ops; support redone per block)
// - X-row trig AND support trig staged in LDS (X pitch 68 -> each lane's
//   b128 window covers banks 4*i+f..+3: 16 disjoint windows, conflict-free;
//   support reads are half-wave broadcasts). Keeps VGPRs < 256: no
//   s_set_vgpr_msb churn, full VALU co-issue.
// - inner math on float2 -> v_pk_fma_f32 / v_pk_mul_f32 (VOP3P packed f32)
// - head contraction sum_j K[i,j]*W'[j] via V_WMMA_F32_16X16X4_F32:
//     A (16x4 f32): lane<16 -> {K[i,jt+0],K[i,jt+1]}, lane>=16 -> {K,jt+2/3}
//     B (4x16 f32): W' replicated across N columns (same half-wave split)
//     C/D (16x16 f32, 8 VGPRs) accumulates over all j tiles
// - cross-wave reduce via ds_add_f32 into LDS, +bias, store.

typedef __attribute__((ext_vector_type(2))) float v2f;
typedef __attribute__((ext_vector_type(4))) float v4f;
typedef __attribute__((ext_vector_type(8))) float v8f;

#define BATCH  2048
#define NSUP   1024
#define NFEAT  64
#define TILEJ  32      // support rows staged in LDS per step
#define BLOCK  256     // 8 waves of 32
#define XPITCH 68      // 64 + 4 pad: disjoint 4-bank windows per lane

__global__ __launch_bounds__(BLOCK) void
qkr_kernel(const float* __restrict__ X,
           const float* __restrict__ Sup,
           const float* __restrict__ W,
           const float* __restrict__ bias,
           float* __restrict__ out)
{
    __shared__ float sW[NSUP];            // W * 2^-64 (0.5 per factor folded)
    __shared__ float sc[TILEJ * NFEAT];   // cos(support tile)
    __shared__ float ss[TILEJ * NFEAT];   // sin(support tile)
    __shared__ float xcL[16 * XPITCH];    // cos(X tile), padded pitch
    __shared__ float xsL[16 * XPITCH];    // sin(X tile), padded pitch
    __shared__ float out_acc[16];

    const int t     = threadIdx.x;
    const int wave  = t >> 5;
    const int lane  = t & 31;
    const int half  = lane >> 4;    // 0: lanes 0-15, 1: lanes 16-31
    const int l16   = lane & 15;
    const int ibase = blockIdx.x * 16;

    // Preload scaled W and zero the cross-wave accumulator.
    for (int j = t; j < NSUP; j += BLOCK) sW[j] = W[j] * 0x1p-64f;
    if (t < 16) out_acc[t] = 0.0f;

    // X tile trig -> LDS (1024 elements, 4 per thread, coalesced reads).
#pragma unroll
    for (int k = 0; k < (16 * NFEAT) / BLOCK; ++k) {
        int idx = t + k * BLOCK;                 // 0..1023
        int i   = idx >> 6;
        int f   = idx & 63;
        float s_, c_;
        __sincosf(X[(ibase + i) * NFEAT + f], &s_, &c_);
        xcL[i * XPITCH + f] = c_;
        xsL[i * XPITCH + f] = s_;
    }

    const float* xcrow = &xcL[l16 * XPITCH];
    const float* xsrow = &xsL[l16 * XPITCH];

    v8f acc = {0.f, 0.f, 0.f, 0.f, 0.f, 0.f, 0.f, 0.f};

#pragma unroll 1
    for (int jb = 0; jb < NSUP; jb += TILEJ) {
        __syncthreads();  // tile reuse fence (covers sW/out_acc/X init too)

        // Cooperative sincos of the support tile into LDS (coalesced reads).
#pragma unroll
        for (int k = 0; k < (TILEJ * NFEAT) / BLOCK; ++k) {
            int idx = t + k * BLOCK;                    // 0..2047
            float s_, c_;
            __sincosf(Sup[(jb + (idx >> 6)) * NFEAT + (idx & 63)], &s_, &c_);
            sc[idx] = c_;
            ss[idx] = s_;
        }
        __syncthreads();

        // This wave's 4 columns within the tile; per lane two j's,
        // laid out to match the 16x4 f32 WMMA A-fragment.
        const int ja  = wave * 4 + half * 2;   // local j for p.x
        const int ja1 = ja + 1;                // local j for p.y
        const float* pca = &sc[ja  * NFEAT];
        const float* psa = &ss[ja  * NFEAT];
        const float* pcb = &sc[ja1 * NFEAT];
        const float* psb = &ss[ja1 * NFEAT];

        v2f p = {1.0f, 1.0f};
#pragma unroll 4
        for (int f = 0; f < NFEAT; f += 4) {
            v4f xcv = *(const v4f*)(xcrow + f);  // conflict-free b128
            v4f xsv = *(const v4f*)(xsrow + f);
            v4f ca  = *(const v4f*)(pca + f);    // broadcast b128
            v4f sa  = *(const v4f*)(psa + f);
            v4f cb  = *(const v4f*)(pcb + f);
            v4f sb  = *(const v4f*)(psb + f);
#pragma unroll
            for (int k = 0; k < 4; ++k) {
                v2f tc   = {ca[k], cb[k]};
                v2f tsn  = {sa[k], sb[k]};
                v2f term = xcv[k] * tc + (xsv[k] * tsn + 1.0f);
                p *= term;                       // v_pk_fma_f32 / v_pk_mul_f32
            }
        }

        // B-fragment: W' values for this wave's 4 rows of B (4x16),
        // replicated across the 16 N columns per half-wave.
        v2f bfrag = { sW[jb + ja], sW[jb + ja1] };

        // D += A(16x4) x B(4x16): accumulates sum_j K[i,j]*W'[j] in every
        // column of the 16x16 f32 accumulator.
        acc = __builtin_amdgcn_wmma_f32_16x16x4_f32(
            false, p, false, bfrag, (short)0, acc, false, false);
    }

    // All 16 columns of D are identical; lane N=0 of each half drains its
    // 8 M-rows (half 0 -> M=0..7, half 1 -> M=8..15) into LDS.
    if (l16 == 0) {
#pragma unroll
        for (int m = 0; m < 8; ++m) {
            atomicAdd(&out_acc[half * 8 + m], acc[m]);   // ds_add_f32
        }
    }
    __syncthreads();

    if (t < 16) out[ibase + t] = out_acc[t] + bias[0];
}

extern "C" void kernel_launch(void* const* d_in, const int* in_sizes, int n_in,
                              void* d_out, int out_size, void* d_ws, size_t ws_size,
                              hipStream_t stream) {
    const float* X   = (const float*)d_in[0];  // [2048, 64]
    const float* Sup = (const float*)d_in[1];  // [1024, 64]
    const float* W   = (const float*)d_in[2];  // [1, 1024]
    const float* b   = (const float*)d_in[3];  // [1]
    float* out = (float*)d_out;                // [2048]

    dim3 grid(BATCH / 16);                     // 128 blocks
    qkr_kernel<<<grid, BLOCK, 0, stream>>>(X, Sup, W, b, out);
}